// myVMLSTMCell_NEO_final_group_33517924778466
// MI455X (gfx1250) — compile-verified
//
#include <hip/hip_runtime.h>
#include <hip/hip_bf16.h>

// MI455X / gfx1250, wave32. Fused low-rank VM-LSTM cell.
// Memory-bound (~180MB @ 23.3TB/s vs ~16 GFLOP) -> use native f32 WMMA
// (V_WMMA_F32_16X16X4_F32) for full precision; fuse expansion + gates so the
// (B x 4H) pre-activation tensors never touch HBM.

typedef __attribute__((ext_vector_type(2))) float v2f;
typedef __attribute__((ext_vector_type(8))) float v8f;

#define NB   8192           // batch
#define HD   1024           // hidden == input dim
// workspace layout (floats)
#define WS_COEFX 0                    // 4*1024
#define WS_COEFH 4096                 // 4*1024
#define WS_TX    8192                 // 8192*64
#define WS_TH0   (8192 + 8192*64)     // 8192*128  (g0 cols 0..63, g1 cols 64..127)
#define WS_TH1   (WS_TH0 + 8192*128)  // 8192*128

__device__ __forceinline__ v8f wmma4(v2f a, v2f b, v8f c) {
  // D = A(16x4 f32) * B(4x16 f32) + C(16x16 f32)
  return __builtin_amdgcn_wmma_f32_16x16x4_f32(
      /*neg_a=*/false, a, /*neg_b=*/false, b,
      /*c_mod=*/(short)0, c, /*reuse_a=*/false, /*reuse_b=*/false);
}

// ---------------------------------------------------------------------------
// Kernel 0: coef_x[j][d] = sum_r Ux[d,r]*Vx[j*H+d,r]
//           coef_h[j][hc] = sum_r Uh0f[hc,r]*Vh0f[j,hc,r]
// ---------------------------------------------------------------------------
__global__ __launch_bounds__(256) void coef_kernel(
    const float* __restrict__ Ux, const float* __restrict__ Vx,
    const float* __restrict__ Uh0, const float* __restrict__ Vh0,
    float* __restrict__ ws) {
  int t = blockIdx.x * blockDim.x + threadIdx.x;
  if (t < 4096) {
    int j = t >> 10, d = t & 1023;
    const float* u = Ux + d * 64;
    const float* v = Vx + (j * 1024 + d) * 64;
    float s = 0.f;
#pragma unroll 8
    for (int r = 0; r < 64; ++r) s += u[r] * v[r];
    ws[WS_COEFX + t] = s;
  } else if (t < 8192) {
    int tt = t - 4096;
    int j = tt >> 10, hc = tt & 1023;
    const float* u = Uh0 + hc * 64;          // Uh_0 flattened (H, r0)
    int f = j * 1024 + hc;                   // flat index into (4H)
    int gf = f >> 11, mf = f & 2047;         // Vh_0 is (2, 64, 2048)
    const float* v = Vh0 + gf * (64 * 2048) + mf;
    float s = 0.f;
#pragma unroll 8
    for (int r = 0; r < 64; ++r) s += u[r] * v[r * 2048];
    ws[WS_COEFH + tt] = s;
  }
}

// ---------------------------------------------------------------------------
// Kernel 1: rank projection.  One wave computes a 16x64 tile of one target:
//   tgt 0: t_x      = x            @ Ux          (K=1024)
//   tgt 1: t_h0[g0] = h[:,  0:512] @ Uh_0[0]     (K=512)
//   tgt 2: t_h0[g1] = h[:,512:1024]@ Uh_0[1]
//   tgt 3: t_h1[g0] = h[:,512:1024]@ Uh_1[0]     (roll by -1)
//   tgt 4: t_h1[g1] = h[:,  0:512] @ Uh_1[1]
// ---------------------------------------------------------------------------
__global__ __launch_bounds__(256) void rank_proj_kernel(
    const float* __restrict__ x, const float* __restrict__ h,
    const float* __restrict__ Ux, const float* __restrict__ Uh0,
    const float* __restrict__ Uh1, float* __restrict__ ws) {
  int wid  = (blockIdx.x * blockDim.x + threadIdx.x) >> 5;  // 0..2559
  int lane = threadIdx.x & 31;
  int rowTile = wid & 511;   // 512 row tiles
  int tgt     = wid >> 9;    // 0..4

  float* t_x  = ws + WS_TX;
  float* t_h0 = ws + WS_TH0;
  float* t_h1 = ws + WS_TH1;

  const float* src; int srcCol, K; const float* W; float* out; int outLd, outCol;
  switch (tgt) {
    default:
    case 0: src = x; srcCol = 0;   K = 1024; W = Ux;             out = t_x;  outLd = 64;  outCol = 0;  break;
    case 1: src = h; srcCol = 0;   K = 512;  W = Uh0;            out = t_h0; outLd = 128; outCol = 0;  break;
    case 2: src = h; srcCol = 512; K = 512;  W = Uh0 + 512 * 64; out = t_h0; outLd = 128; outCol = 64; break;
    case 3: src = h; srcCol = 512; K = 512;  W = Uh1;            out = t_h1; outLd = 128; outCol = 0;  break;
    case 4: src = h; srcCol = 0;   K = 512;  W = Uh1 + 512 * 64; out = t_h1; outLd = 128; outCol = 64; break;
  }

  int m  = lane & 15;   // A row / B col within tile
  int kh = lane >> 4;   // selects K pair {2*kh, 2*kh+1}

  const float* aPtr = src + (size_t)(rowTile * 16 + m) * HD + srcCol + 2 * kh;
  const float* bPtr = W + (2 * kh) * 64 + m;  // W[(k)*64 + n]

  v8f acc0 = {0,0,0,0,0,0,0,0}, acc1 = {0,0,0,0,0,0,0,0};
  v8f acc2 = {0,0,0,0,0,0,0,0}, acc3 = {0,0,0,0,0,0,0,0};

  for (int kk = 0; kk < K; kk += 4) {
    v2f a = *(const v2f*)(aPtr + kk);
    const float* bp = bPtr + kk * 64;
    v2f b0 = { bp[0],  bp[64]  };
    v2f b1 = { bp[16], bp[80]  };
    v2f b2 = { bp[32], bp[96]  };
    v2f b3 = { bp[48], bp[112] };
    acc0 = wmma4(a, b0, acc0);
    acc1 = wmma4(a, b1, acc1);
    acc2 = wmma4(a, b2, acc2);
    acc3 = wmma4(a, b3, acc3);
  }

  float* o = out + (size_t)(rowTile * 16 + 8 * kh) * outLd + outCol + m;
#pragma unroll
  for (int v = 0; v < 8; ++v) {
    o[v * outLd + 0]  = acc0[v];
    o[v * outLd + 16] = acc1[v];
    o[v * outLd + 32] = acc2[v];
    o[v * outLd + 48] = acc3[v];
  }
}

// ---------------------------------------------------------------------------
// Kernel 2: rank expansion (K=64) + full gate fusion.
// One wave handles a 16-row x 16-hidden-col tile: 4 gate accumulators,
// 48 f32 WMMAs per gate tile, then sigmoid/tanh cell update.
// Gate pairing: input=(jx0,jh1) forget=(jx1,jh0) output=(jx2,jh3) new=(jx3,jh2)
// ---------------------------------------------------------------------------
__global__ __launch_bounds__(256) void expand_gate_kernel(
    const float* __restrict__ x, const float* __restrict__ h,
    const float* __restrict__ c,
    const float* __restrict__ dia_x, const float* __restrict__ dia_h,
    const float* __restrict__ Vx,
    const float* __restrict__ Vh0, const float* __restrict__ Vh1,
    const float* __restrict__ bias_x, const float* __restrict__ bias_h,
    const float* __restrict__ ws, float* __restrict__ out) {
  int wid  = (blockIdx.x * blockDim.x + threadIdx.x) >> 5;  // 0..32767
  int lane = threadIdx.x & 31;
  int rowTile = wid >> 6;        // 0..511
  int hcTile  = wid & 63;        // 0..63
  int rowBase = rowTile * 16;
  int hc0     = hcTile * 16;

  const float* coef_x = ws + WS_COEFX;
  const float* coef_h = ws + WS_COEFH;
  const float* t_x    = ws + WS_TX;
  const float* t_h0   = ws + WS_TH0;
  const float* t_h1   = ws + WS_TH1;

  int n  = lane & 15;   // A row (M) == B/C col (N) for this lane
  int kh = lane >> 4;

  // Resident A-fragment for t_x (16 K-steps of 4)
  v2f ax[16];
  {
    const float* p = t_x + (size_t)(rowBase + n) * 64 + 2 * kh;
#pragma unroll
    for (int s = 0; s < 16; ++s) ax[s] = *(const v2f*)(p + 4 * s);
  }

  v2f ah0[16], ah1[16];
  auto loadAH = [&](int g) {
    const float* p0 = t_h0 + (size_t)(rowBase + n) * 128 + g * 64 + 2 * kh;
    const float* p1 = t_h1 + (size_t)(rowBase + n) * 128 + g * 64 + 2 * kh;
#pragma unroll
    for (int s = 0; s < 16; ++s) {
      ah0[s] = *(const v2f*)(p0 + 4 * s);
      ah1[s] = *(const v2f*)(p1 + 4 * s);
    }
  };

  auto computeGate = [&](int jx, int jh) -> v8f {
    v8f a0 = {0,0,0,0,0,0,0,0}, a1 = {0,0,0,0,0,0,0,0}, a2 = {0,0,0,0,0,0,0,0};
    const float* pVx = Vx + (size_t)(jx * 1024 + hc0 + n) * 64 + 2 * kh;
    int g  = jh >> 1;
    int m0 = (jh & 1) * 1024 + hc0;
    const float* pV0 = Vh0 + (size_t)(g * 64 + 2 * kh) * 2048 + m0 + n;
    const float* pV1 = Vh1 + (size_t)(g * 64 + 2 * kh) * 2048 + m0 + n;
#pragma unroll
    for (int s = 0; s < 16; ++s) {
      v2f bx = *(const v2f*)(pVx + 4 * s);
      a0 = wmma4(ax[s], bx, a0);
      v2f b0 = { pV0[s * 8192], pV0[s * 8192 + 2048] };
      a1 = wmma4(ah0[s], b0, a1);
      v2f b1 = { pV1[s * 8192], pV1[s * 8192 + 2048] };
      a2 = wmma4(ah1[s], b1, a2);
    }
    return a0 + a1 + a2;
  };

  loadAH(0);                       // group 0 serves jh in {0,1}
  v8f accI = computeGate(0, 1);    // inputgate : xi + hi
  v8f accF = computeGate(1, 0);    // forgetgate: xf + hf
  loadAH(1);                       // group 1 serves jh in {2,3}
  v8f accO = computeGate(2, 3);    // outputgate: xo + ho
  v8f accN = computeGate(3, 2);    // newgate   : xn + hn

  // Elementwise fusion
  int hc = hc0 + n;
  float dx  = dia_x[hc],            dh  = dia_h[hc];
  float cx0 = coef_x[hc],           cx1 = coef_x[1024 + hc];
  float cx2 = coef_x[2048 + hc],    cx3 = coef_x[3072 + hc];
  float ch0 = coef_h[hc],           ch1 = coef_h[1024 + hc];
  float ch2 = coef_h[2048 + hc],    ch3 = coef_h[3072 + hc];
  float bx0 = bias_x[hc],           bx1 = bias_x[1024 + hc];
  float bx2 = bias_x[2048 + hc],    bx3 = bias_x[3072 + hc];
  float bh0 = bias_h[hc],           bh1 = bias_h[1024 + hc];
  float bh2 = bias_h[2048 + hc],    bh3 = bias_h[3072 + hc];

  float* out_h = out;
  float* out_c = out + (size_t)NB * HD;

#pragma unroll
  for (int v = 0; v < 8; ++v) {
    size_t r = (size_t)(rowBase + 8 * kh + v);
    size_t idx = r * HD + hc;
    float xv = x[idx], hv = h[idx], cv = c[idx];
    float base = dx * xv + dh * hv;
    float preI = accI[v] - xv * cx0 + bx0 - hv * ch1 + bh1 + base;
    float preF = accF[v] - xv * cx1 + bx1 - hv * ch0 + bh0 + base;
    float preO = accO[v] - xv * cx2 + bx2 - hv * ch3 + bh3 + base;
    float preN = accN[v] - xv * cx3 + bx3 - hv * ch2 + bh2 + base;
    float ig = 1.f / (1.f + __expf(-preI));
    float fg = 1.f / (1.f + __expf(-preF));
    float og = 1.f / (1.f + __expf(-preO));
    float ng = tanhf(preN);
    float cn = fg * cv + ig * ng;
    float hn = og * tanhf(cn);
    out_h[idx] = hn;
    out_c[idx] = cn;
  }
}

// ---------------------------------------------------------------------------
extern "C" void kernel_launch(void* const* d_in, const int* in_sizes, int n_in,
                              void* d_out, int out_size, void* d_ws, size_t ws_size,
                              hipStream_t stream) {
  const float* x      = (const float*)d_in[0];
  const float* h      = (const float*)d_in[1];
  const float* c      = (const float*)d_in[2];
  const float* dia_x  = (const float*)d_in[3];
  const float* dia_h  = (const float*)d_in[4];
  const float* Ux     = (const float*)d_in[5];
  const float* Vx     = (const float*)d_in[6];
  const float* Uh0    = (const float*)d_in[7];
  const float* Vh0    = (const float*)d_in[8];
  const float* Uh1    = (const float*)d_in[9];
  const float* Vh1    = (const float*)d_in[10];
  const float* bias_x = (const float*)d_in[11];
  const float* bias_h = (const float*)d_in[12];
  float* out = (float*)d_out;
  float* ws  = (float*)d_ws;

  // 8192 coef threads
  coef_kernel<<<32, 256, 0, stream>>>(Ux, Vx, Uh0, Vh0, ws);
  // 5 targets x 512 row tiles = 2560 waves -> 320 blocks of 8 waves
  rank_proj_kernel<<<320, 256, 0, stream>>>(x, h, Ux, Uh0, Uh1, ws);
  // 512 row tiles x 64 col tiles = 32768 waves -> 4096 blocks
  expand_gate_kernel<<<4096, 256, 0, stream>>>(x, h, c, dia_x, dia_h, Vx, Vh0,
                                               Vh1, bias_x, bias_h, ws, out);
}